// MultimodalAttention_26456998543981
// MI455X (gfx1250) — compile-verified
//
#include <hip/hip_runtime.h>
#include <hip/hip_bf16.h>
#include <math.h>

// ---------------------------------------------------------------------------
// MultimodalAttention for MI455X (gfx1250, wave32)
// B=64, L=1024, LI=256, H=512, A=1024
// Memory-bound (~670MB streamed); GEMMs done in fp32 WMMA (v_wmma_f32_16x16x4_f32)
// ---------------------------------------------------------------------------

typedef __attribute__((ext_vector_type(2))) float v2f;
typedef __attribute__((ext_vector_type(8))) float v8f;

#define A_DIM 1024
#define B_DIM 64
#define L_TXT 1024
#define L_IMG 256

// ---------------------------------------------------------------------------
// dec = concat(dec_hidden_0, dec_hidden_1)  -> (64, 1024)
// ---------------------------------------------------------------------------
__global__ __launch_bounds__(256)
void mm_concat_kernel(const float* __restrict__ h0, const float* __restrict__ h1,
                      float* __restrict__ dec) {
    int i = blockIdx.x * 256 + threadIdx.x;          // 0 .. 65535
    int b = i >> 10;
    int k = i & 1023;
    dec[i] = (k < 512) ? h0[b * 512 + k] : h1[b * 512 + (k - 512)];
}

// ---------------------------------------------------------------------------
// D[M,N] = X[M,K] @ W[K,N] (+ bias[N]) (+ Cin[M,N])
// One wave computes one 16x16 output tile via V_WMMA_F32_16X16X4_F32.
// 8 waves / block, grid = (N/16/8, M/16).
//
// f32 A-frag (16x4): lane<16 -> M=lane, K={0,1}; lane>=16 -> M=lane-16, K={2,3}
// f32 B-frag (4x16): lane<16 -> N=lane, K={0,1}; lane>=16 -> N=lane-16, K={2,3}
// C/D (16x16): vgpr i, lanes0-15 -> M=i, N=lane; lanes16-31 -> M=i+8, N=lane-16
// ---------------------------------------------------------------------------
__global__ __launch_bounds__(256)
void mm_gemm_wmma_f32(const float* __restrict__ X, const float* __restrict__ W,
                      const float* __restrict__ bias, const float* __restrict__ Cin,
                      float* __restrict__ D, int N, int K) {
    const int wave = threadIdx.x >> 5;
    const int lane = threadIdx.x & 31;
    const int tn   = (blockIdx.x * 8 + wave) * 16;
    const int tm   = blockIdx.y * 16;

    const int lo   = lane & 15;
    const int koff = (lane >> 4) * 2;      // 0 for lanes 0-15, 2 for lanes 16-31

    const float* xr = X + (size_t)(tm + lo) * K;   // A row for this lane
    const int    nc = tn + lo;                     // B/C column for this lane

    v8f c = {};
    for (int kk = 0; kk < K; kk += 4) {
        v2f a, b;
        a.x = xr[kk + koff];
        a.y = xr[kk + koff + 1];
        const float* wr = W + (size_t)(kk + koff) * N + nc;
        b.x = wr[0];
        b.y = wr[N];
        c = __builtin_amdgcn_wmma_f32_16x16x4_f32(
                /*neg_a=*/false, a, /*neg_b=*/false, b,
                /*c_mod=*/(short)0, c, /*reuse_a=*/false, /*reuse_b=*/false);
    }

    const int row0 = tm + (lane >> 4) * 8;
#pragma unroll
    for (int i = 0; i < 8; ++i) {
        float r = c[i];
        if (bias) r += bias[nc];
        if (Cin)  r += Cin[(size_t)(row0 + i) * N + nc];
        D[(size_t)(row0 + i) * N + nc] = r;
    }
}

// ---------------------------------------------------------------------------
// e[b,l] = sum_a v[a] * tanh(feat[b,l,a] + df[b,a] + coverage_set*cov[b,l]*w[a])
// One wave per (b,l) row; lane-strided over A (coalesced 128B bursts).
// ---------------------------------------------------------------------------
__global__ __launch_bounds__(256)
void mm_escore_kernel(const float* __restrict__ feat, const float* __restrict__ df,
                      const float* __restrict__ cov, const float* __restrict__ w,
                      const float* __restrict__ v, const int* __restrict__ cs,
                      float* __restrict__ e, int Lr) {
    const int wave = threadIdx.x >> 5;
    const int lane = threadIdx.x & 31;
    const int r    = blockIdx.x * 8 + wave;        // row in [0, B*Lr)
    const int b    = r / Lr;

    const bool  use_cov = (*cs != 0);
    const float cv      = use_cov ? cov[r] : 0.0f;
    const float* f = feat + (size_t)r * A_DIM;
    const float* d = df   + (size_t)b * A_DIM;

    float s = 0.0f;
    for (int j = lane; j < A_DIM; j += 32) {
        float x = f[j] + d[j] + cv * w[j];
        s += v[j] * tanhf(x);
    }
#pragma unroll
    for (int off = 16; off; off >>= 1) s += __shfl_xor(s, off, 32);
    if (lane == 0) e[r] = s;
}

// ---------------------------------------------------------------------------
// attn = softmax(e) (*mask, renormalized) ; cov_out = cov_in + attn (or attn)
// Algebraic fusion: softmax*mask renormalized == exp(e-m)*mask / sum(exp*mask)
// One 256-thread block per batch row.
// ---------------------------------------------------------------------------
__global__ __launch_bounds__(256)
void mm_softmax_cov_kernel(const float* __restrict__ e, const float* __restrict__ mask,
                           const float* __restrict__ cov_in, const int* __restrict__ cs,
                           float* __restrict__ attn, float* __restrict__ cov_out, int n) {
    __shared__ float sred[8];
    const int b    = blockIdx.x;
    const int tid  = threadIdx.x;
    const int lane = tid & 31, wv = tid >> 5;
    const float* er = e + (size_t)b * n;

    float m = -3.402823466e38f;
    for (int j = tid; j < n; j += 256) m = fmaxf(m, er[j]);
#pragma unroll
    for (int off = 16; off; off >>= 1) m = fmaxf(m, __shfl_xor(m, off, 32));
    if (lane == 0) sred[wv] = m;
    __syncthreads();
    float mm = sred[0];
#pragma unroll
    for (int i = 1; i < 8; ++i) mm = fmaxf(mm, sred[i]);
    __syncthreads();

    float s = 0.0f;
    for (int j = tid; j < n; j += 256) {
        float p = __expf(er[j] - mm);
        if (mask) p *= mask[(size_t)b * n + j];
        s += p;
    }
#pragma unroll
    for (int off = 16; off; off >>= 1) s += __shfl_xor(s, off, 32);
    if (lane == 0) sred[wv] = s;
    __syncthreads();
    float ss = 0.0f;
#pragma unroll
    for (int i = 0; i < 8; ++i) ss += sred[i];
    const float inv = 1.0f / ss;

    const bool add_cov = (*cs != 0);
    for (int j = tid; j < n; j += 256) {
        float p = __expf(er[j] - mm);
        if (mask) p *= mask[(size_t)b * n + j];
        float a = p * inv;
        attn[(size_t)b * n + j]    = a;
        cov_out[(size_t)b * n + j] = add_cov ? (cov_in[(size_t)b * n + j] + a) : a;
    }
}

// ---------------------------------------------------------------------------
// ctx[b,a] = sum_l attn[b,l] * enc[b,l,a]
// attn row staged in LDS; each thread owns one output column (coalesced).
// grid = (A/256, B)
// ---------------------------------------------------------------------------
__global__ __launch_bounds__(256)
void mm_ctx_kernel(const float* __restrict__ attn, const float* __restrict__ enc,
                   float* __restrict__ ctx, int Lr) {
    __shared__ float sa[L_TXT];
    const int b = blockIdx.y;
    const int a = blockIdx.x * 256 + threadIdx.x;

    for (int j = threadIdx.x; j < Lr; j += 256) sa[j] = attn[(size_t)b * Lr + j];
    __syncthreads();

    const float* base = enc + (size_t)b * Lr * A_DIM + a;
    float acc = 0.0f;
#pragma unroll 4
    for (int l = 0; l < Lr; ++l) acc += sa[l] * base[(size_t)l * A_DIM];
    ctx[(size_t)b * A_DIM + a] = acc;
}

// ---------------------------------------------------------------------------
// ctx_mm = v3*tanh(g1)*ctx_t + v4*tanh(g2)*ctx_i   (g already holds Xh + ctx@W)
// ---------------------------------------------------------------------------
__global__ __launch_bounds__(256)
void mm_final_kernel(const float* __restrict__ g1, const float* __restrict__ g2,
                     const float* __restrict__ ctx_t, const float* __restrict__ ctx_i,
                     const float* __restrict__ v3, const float* __restrict__ v4,
                     float* __restrict__ out) {
    int i = blockIdx.x * 256 + threadIdx.x;          // 0 .. 65535
    int a = i & 1023;
    float beta1 = v3[a] * tanhf(g1[i]);
    float beta2 = v4[a] * tanhf(g2[i]);
    out[i] = beta1 * ctx_t[i] + beta2 * ctx_i[i];
}

// ---------------------------------------------------------------------------
extern "C" void kernel_launch(void* const* d_in, const int* in_sizes, int n_in,
                              void* d_out, int out_size, void* d_ws, size_t ws_size,
                              hipStream_t stream) {
    const float* dh0          = (const float*)d_in[0];
    const float* dh1          = (const float*)d_in[1];
    const float* enc_outputs  = (const float*)d_in[2];
    const float* enc_features = (const float*)d_in[3];
    const float* enc_mask     = (const float*)d_in[4];
    const float* coverage     = (const float*)d_in[5];
    const float* enc_img_out  = (const float*)d_in[6];
    const float* enc_img_feat = (const float*)d_in[7];
    /* d_in[8] img_mask is unused by the reference's attn_img path */
    const float* coverage_img = (const float*)d_in[9];
    const int*   cs           = (const int*)d_in[10];
    const float* W1 = (const float*)d_in[11]; const float* b1 = (const float*)d_in[12];
    const float* W2 = (const float*)d_in[13]; const float* b2 = (const float*)d_in[14];
    const float* W3 = (const float*)d_in[15]; const float* b3 = (const float*)d_in[16];
    const float* W4 = (const float*)d_in[17]; const float* b4 = (const float*)d_in[18];
    const float* WA = (const float*)d_in[19]; const float* WB = (const float*)d_in[20];
    const float* v1 = (const float*)d_in[21]; const float* v2 = (const float*)d_in[22];
    const float* v3 = (const float*)d_in[23]; const float* v4 = (const float*)d_in[24];
    const float* wc = (const float*)d_in[25]; const float* wci = (const float*)d_in[26];

    // workspace layout (floats)
    float* ws    = (float*)d_ws;
    float* dec   = ws;                  // 65536
    float* h1    = ws + 65536;          // df   = dec@W1+b1
    float* h2    = ws + 131072;         // dfi  = dec@W2+b2
    float* h3    = ws + 196608;         // dec@W3+b3
    float* h4    = ws + 262144;         // dec@W4+b4
    float* e     = ws + 327680;         // 65536
    float* eimg  = ws + 393216;         // 16384
    float* ctx_t = ws + 409600;         // 65536
    float* ctx_i = ws + 475136;         // 65536
    float* g1    = ws + 540672;         // 65536
    float* g2    = ws + 606208;         // 65536  (end: 671744 floats = 2.7MB)

    // output layout: ctx_mm | attn | attn_img | coverage | coverage_img
    float* out          = (float*)d_out;
    float* ctx_mm       = out;
    float* attn         = out + 65536;
    float* attn_img     = out + 131072;
    float* cov_out      = out + 147456;
    float* cov_img_out  = out + 212992;

    // 1) dec = concat(h0, h1)
    mm_concat_kernel<<<256, 256, 0, stream>>>(dh0, dh1, dec);

    // 2) four WMMA GEMMs: h_k = dec @ W_k + b_k   (M=64, N=1024, K=1024)
    dim3 gg(A_DIM / 16 / 8, B_DIM / 16);
    mm_gemm_wmma_f32<<<gg, 256, 0, stream>>>(dec, W1, b1, nullptr, h1, A_DIM, A_DIM);
    mm_gemm_wmma_f32<<<gg, 256, 0, stream>>>(dec, W2, b2, nullptr, h2, A_DIM, A_DIM);
    mm_gemm_wmma_f32<<<gg, 256, 0, stream>>>(dec, W3, b3, nullptr, h3, A_DIM, A_DIM);
    mm_gemm_wmma_f32<<<gg, 256, 0, stream>>>(dec, W4, b4, nullptr, h4, A_DIM, A_DIM);

    // 3) attention energies (the 335MB streaming part)
    mm_escore_kernel<<<(B_DIM * L_TXT) / 8, 256, 0, stream>>>(
        enc_features, h1, coverage, wc, v1, cs, e, L_TXT);
    mm_escore_kernel<<<(B_DIM * L_IMG) / 8, 256, 0, stream>>>(
        enc_img_feat, h2, coverage_img, wci, v2, cs, eimg, L_IMG);

    // 4) softmax (+mask renorm fused) + coverage update
    mm_softmax_cov_kernel<<<B_DIM, 256, 0, stream>>>(
        e, enc_mask, coverage, cs, attn, cov_out, L_TXT);
    mm_softmax_cov_kernel<<<B_DIM, 256, 0, stream>>>(
        eimg, nullptr, coverage_img, cs, attn_img, cov_img_out, L_IMG);

    // 5) contexts (the other 335MB streaming part)
    mm_ctx_kernel<<<dim3(A_DIM / 256, B_DIM), 256, 0, stream>>>(attn, enc_outputs, ctx_t, L_TXT);
    mm_ctx_kernel<<<dim3(A_DIM / 256, B_DIM), 256, 0, stream>>>(attn_img, enc_img_out, ctx_i, L_IMG);

    // 6) g1 = ctx_t @ WA + h3 ; g2 = ctx_i @ WB + h4 (WMMA, additive C-input)
    mm_gemm_wmma_f32<<<gg, 256, 0, stream>>>(ctx_t, WA, nullptr, h3, g1, A_DIM, A_DIM);
    mm_gemm_wmma_f32<<<gg, 256, 0, stream>>>(ctx_i, WB, nullptr, h4, g2, A_DIM, A_DIM);

    // 7) ctx_mm = v3*tanh(g1)*ctx_t + v4*tanh(g2)*ctx_i
    mm_final_kernel<<<(B_DIM * A_DIM) / 256, 256, 0, stream>>>(
        g1, g2, ctx_t, ctx_i, v3, v4, ctx_mm);
}